// EdgeGNNClassifier_56624848830942
// MI455X (gfx1250) — compile-verified
//
#include <hip/hip_runtime.h>
#include <hip/hip_bf16.h>

// ---------------------------------------------------------------------------
// EdgeGNN (2x GINEConv + edge MLP) for gfx1250.
// All GEMMs use V_WMMA_F32_16X16X4_F32 (full fp32, wave32 matrix path).
// Problem is L2/scatter bound (~330MB HBM floor), so f32 WMMA costs nothing
// and keeps numerics identical to the reference.
// ---------------------------------------------------------------------------

typedef float v2f __attribute__((ext_vector_type(2)));
typedef float v8f __attribute__((ext_vector_type(8)));

#define N_NODES 50000
#define N_EDGES 800000

// One wave computes a 16xK * Kx64 tile: D[16][64] = A[16][K] * W[K][64] + bias.
// A: per-wave LDS tile, row stride K+1 (bank-conflict padding).
// W: block-shared LDS, row-major KxN (stride 64).
// Result: acc[t][v] = D[v + 8*(lane>>4)][t*16 + (lane&15)]  (ISA C/D layout).
template <int K>
__device__ __forceinline__ void wave_gemm_16xK_64(const float* A, const float* W,
                                                  const float* Bias, v8f (&acc)[4],
                                                  int lane) {
  const int c15  = lane & 15;   // A: M row / B: N col held by this lane
  const int half = lane >> 4;   // selects K pair within each WMMA step
  constexpr int LDA = K + 1;
#pragma unroll
  for (int t = 0; t < 4; ++t) {
    const float bv = Bias[t * 16 + c15];
    v8f c = {bv, bv, bv, bv, bv, bv, bv, bv};
#pragma unroll
    for (int kk = 0; kk < K / 4; ++kk) {
      const int kb = kk * 4 + half * 2;
      v2f a;
      a.x = A[c15 * LDA + kb];
      a.y = A[c15 * LDA + kb + 1];
      v2f b;
      b.x = W[kb * 64 + t * 16 + c15];
      b.y = W[(kb + 1) * 64 + t * 16 + c15];
      c = __builtin_amdgcn_wmma_f32_16x16x4_f32(false, a, false, b, (short)0, c,
                                                false, false);
    }
    acc[t] = c;
  }
}

// ---------------------------------------------------------------------------
__global__ void zero_f32_kernel(float* __restrict__ p, int n) {
  int i = blockIdx.x * blockDim.x + threadIdx.x;
  if (i < n) p[i] = 0.0f;
}

// ---------------------------------------------------------------------------
// GINE edge stage: e = edge_attr @ We + be   (WMMA, K=32)
//                  msg = relu(xin[src] + e); atomicAdd(agg[dst], msg)
// One wave = 16 edges. 8 waves / block.
__global__ void __launch_bounds__(256)
edge_conv_kernel(const float* __restrict__ ea, const float* __restrict__ We,
                 const float* __restrict__ be, const long long* __restrict__ ei,
                 const float* __restrict__ xin, float* __restrict__ agg) {
  __shared__ float sW[32 * 64];
  __shared__ float sB[64];
  __shared__ float sA[8][16 * 33];  // padded stride 33

  const int tid = threadIdx.x, wv = tid >> 5, lane = tid & 31;
  for (int i = tid; i < 32 * 64; i += 256) sW[i] = We[i];
  if (tid < 64) sB[tid] = be[tid];
  __syncthreads();

  const int e0 = (blockIdx.x * 8 + wv) * 16;
  float* A = sA[wv];
  for (int i = lane; i < 16 * 32; i += 32) {
    int r = i >> 5, c = i & 31;
    int e = e0 + r;
    e = e < N_EDGES ? e : N_EDGES - 1;
    A[r * 33 + c] = ea[e * 32 + c];
  }
  __syncthreads();

  v8f acc[4];
  wave_gemm_16xK_64<32>(A, sW, sB, acc, lane);

  const long long* srcp = ei;
  const long long* dstp = ei + N_EDGES;
  const int half = lane >> 4, c15 = lane & 15;

  long long sIdx[8], dIdx[8];
#pragma unroll
  for (int v = 0; v < 8; ++v) {
    int e = e0 + v + 8 * half;
    bool ok = e < N_EDGES;
    int ec = ok ? e : 0;
    sIdx[v] = srcp[ec];
    dIdx[v] = ok ? dstp[ec] : (long long)-1;
  }
#pragma unroll
  for (int t = 0; t < 4; ++t) {
#pragma unroll
    for (int v = 0; v < 8; ++v) {
      if (dIdx[v] >= 0) {
        int col = t * 16 + c15;
        float m = xin[sIdx[v] * 64 + col] + acc[t][v];
        m = fmaxf(m, 0.0f);
        atomicAdd(&agg[dIdx[v] * 64 + col], m);
      }
    }
  }
}

// ---------------------------------------------------------------------------
// Node MLP: out = relu( relu((agg + xin) @ Wa + ba) @ Wb + bb )
// Two chained K=64 WMMA GEMMs; intermediate restaged via per-wave LDS tile.
__global__ void __launch_bounds__(256)
node_mlp_kernel(const float* __restrict__ agg, const float* __restrict__ xin,
                const float* __restrict__ Wa, const float* __restrict__ ba,
                const float* __restrict__ Wb, const float* __restrict__ bb,
                float* __restrict__ out) {
  __shared__ float sWa[64 * 64];
  __shared__ float sWb[64 * 64];
  __shared__ float sBa[64];
  __shared__ float sBb[64];
  __shared__ float sA[8][16 * 65];  // padded stride 65

  const int tid = threadIdx.x, wv = tid >> 5, lane = tid & 31;
  for (int i = tid; i < 64 * 64; i += 256) {
    sWa[i] = Wa[i];
    sWb[i] = Wb[i];
  }
  if (tid < 64) {
    sBa[tid] = ba[tid];
    sBb[tid] = bb[tid];
  }
  __syncthreads();

  const int n0 = (blockIdx.x * 8 + wv) * 16;
  float* A = sA[wv];
  for (int i = lane; i < 16 * 64; i += 32) {
    int r = i >> 6, c = i & 63;
    int n = n0 + r;
    n = n < N_NODES ? n : N_NODES - 1;
    A[r * 65 + c] = agg[n * 64 + c] + xin[n * 64 + c];  // (1+eps)*x, eps=0
  }
  __syncthreads();

  v8f acc[4];
  wave_gemm_16xK_64<64>(A, sWa, sBa, acc, lane);

  const int half = lane >> 4, c15 = lane & 15;
#pragma unroll
  for (int t = 0; t < 4; ++t)
#pragma unroll
    for (int v = 0; v < 8; ++v)
      A[(v + 8 * half) * 65 + t * 16 + c15] = fmaxf(acc[t][v], 0.0f);
  __syncthreads();

  wave_gemm_16xK_64<64>(A, sWb, sBb, acc, lane);

#pragma unroll
  for (int t = 0; t < 4; ++t)
#pragma unroll
    for (int v = 0; v < 8; ++v) {
      int n = n0 + v + 8 * half;
      if (n < N_NODES) out[n * 64 + t * 16 + c15] = fmaxf(acc[t][v], 0.0f);
    }
}

// ---------------------------------------------------------------------------
// Edge MLP: out[e] = relu(concat(h[src[e]], ea[e]) @ Wm1 + bm1) @ Wm2 + bm2
// K=96 WMMA GEMM, then 64->1 projection as a per-row VALU dot product.
__global__ void __launch_bounds__(256)
edge_mlp_kernel(const float* __restrict__ h, const float* __restrict__ ea,
                const long long* __restrict__ ei, const float* __restrict__ Wm1,
                const float* __restrict__ bm1, const float* __restrict__ Wm2,
                const float* __restrict__ bm2, float* __restrict__ out) {
  __shared__ float sW[96 * 64];  // 24 KB
  __shared__ float sB[64];
  __shared__ float sW2[64];
  __shared__ float sA[8][16 * 97];  // padded stride 97, ~50 KB
  __shared__ long long sSrc[8][16];

  const int tid = threadIdx.x, wv = tid >> 5, lane = tid & 31;
  for (int i = tid; i < 96 * 64; i += 256) sW[i] = Wm1[i];
  if (tid < 64) {
    sB[tid] = bm1[tid];
    sW2[tid] = Wm2[tid];
  }
  const int e0 = (blockIdx.x * 8 + wv) * 16;
  if (lane < 16) {
    int e = e0 + lane;
    sSrc[wv][lane] = ei[e < N_EDGES ? e : N_EDGES - 1];
  }
  __syncthreads();

  float* A = sA[wv];
  for (int i = lane; i < 16 * 96; i += 32) {
    int r = i / 96, c = i - r * 96;
    int e = e0 + r;
    e = e < N_EDGES ? e : N_EDGES - 1;
    A[r * 97 + c] = (c < 64) ? h[sSrc[wv][r] * 64 + c] : ea[e * 32 + (c - 64)];
  }
  __syncthreads();

  v8f acc[4];
  wave_gemm_16xK_64<96>(A, sW, sB, acc, lane);

  const int half = lane >> 4, c15 = lane & 15;
#pragma unroll
  for (int t = 0; t < 4; ++t)
#pragma unroll
    for (int v = 0; v < 8; ++v)
      A[(v + 8 * half) * 65 + t * 16 + c15] = fmaxf(acc[t][v], 0.0f);
  __syncthreads();

  if (lane < 16) {
    int e = e0 + lane;
    if (e < N_EDGES) {
      float d = bm2[0];
#pragma unroll
      for (int c = 0; c < 64; ++c) d += A[lane * 65 + c] * sW2[c];
      out[e] = d;  // final layer: no relu
    }
  }
}

// ---------------------------------------------------------------------------
extern "C" void kernel_launch(void* const* d_in, const int* in_sizes, int n_in,
                              void* d_out, int out_size, void* d_ws,
                              size_t ws_size, hipStream_t stream) {
  const float* x = (const float*)d_in[0];
  const long long* ei = (const long long*)d_in[1];
  const float* ea = (const float*)d_in[2];
  const float* We1 = (const float*)d_in[3];
  const float* be1 = (const float*)d_in[4];
  const float* W1a = (const float*)d_in[5];
  const float* b1a = (const float*)d_in[6];
  const float* W1b = (const float*)d_in[7];
  const float* b1b = (const float*)d_in[8];
  const float* We2 = (const float*)d_in[9];
  const float* be2 = (const float*)d_in[10];
  const float* W2a = (const float*)d_in[11];
  const float* b2a = (const float*)d_in[12];
  const float* W2b = (const float*)d_in[13];
  const float* b2b = (const float*)d_in[14];
  const float* Wm1 = (const float*)d_in[15];
  const float* bm1 = (const float*)d_in[16];
  const float* Wm2 = (const float*)d_in[17];
  const float* bm2 = (const float*)d_in[18];
  float* out = (float*)d_out;

  // Workspace: agg (reused for both convs), h1, h2 : each N*64 f32.
  float* agg = (float*)d_ws;
  float* h1 = agg + (size_t)N_NODES * 64;
  float* h2 = h1 + (size_t)N_NODES * 64;

  const int nodeElems = N_NODES * 64;
  const int zeroBlocks = (nodeElems + 255) / 256;
  const int edgeBlocks = (N_EDGES + 127) / 128;  // 16 edges/wave * 8 waves
  const int nodeBlocks = (N_NODES + 127) / 128;  // 16 nodes/wave * 8 waves

  // ---- conv1 ----
  zero_f32_kernel<<<zeroBlocks, 256, 0, stream>>>(agg, nodeElems);
  edge_conv_kernel<<<edgeBlocks, 256, 0, stream>>>(ea, We1, be1, ei, x, agg);
  node_mlp_kernel<<<nodeBlocks, 256, 0, stream>>>(agg, x, W1a, b1a, W1b, b1b, h1);

  // ---- conv2 ----
  zero_f32_kernel<<<zeroBlocks, 256, 0, stream>>>(agg, nodeElems);
  edge_conv_kernel<<<edgeBlocks, 256, 0, stream>>>(ea, We2, be2, ei, h1, agg);
  node_mlp_kernel<<<nodeBlocks, 256, 0, stream>>>(agg, h1, W2a, b2a, W2b, b2b, h2);

  // ---- edge MLP ----
  edge_mlp_kernel<<<edgeBlocks, 256, 0, stream>>>(h2, ea, ei, Wm1, bm1, Wm2, bm2,
                                                  out);
}